// NewRnn_38912403702233
// MI455X (gfx1250) — compile-verified
//
#include <hip/hip_runtime.h>
#include <hip/hip_bf16.h>

#define HDIM 256
#define TLEN 1024

typedef __attribute__((ext_vector_type(2))) float v2f;
typedef __attribute__((ext_vector_type(8))) float v8f;

// ---------------------------------------------------------------------------
// Kernel 1: bulk pass-through copy of the embedding table (HBM-bound, float4)
// ---------------------------------------------------------------------------
__global__ __launch_bounds__(256) void emb_copy_kernel(
    const float* __restrict__ src, float* __restrict__ dst, long n4) {
  long i = (long)blockIdx.x * blockDim.x + threadIdx.x;
  long stride = (long)gridDim.x * blockDim.x;
  const float4* s = (const float4*)src;
  float4* d = (float4*)dst;
  for (; i < n4; i += stride) d[i] = s[i];
}

// ---------------------------------------------------------------------------
// Kernel 2: xp[t,n] = sum_k emb[idx[t],k] * Wih[n,k] + bih[n] + bhh[n]
// One wave per 16x16 tile, V_WMMA_F32_16X16X4_F32, K accumulated in steps of 4.
// Result staged into out[idx[t]] (those rows are overwritten by kernel 3).
// ---------------------------------------------------------------------------
__global__ __launch_bounds__(128) void xproj_wmma_kernel(
    const float* __restrict__ emb, const float* __restrict__ Wih,
    const float* __restrict__ bih, const float* __restrict__ bhh,
    const long long* __restrict__ idx, float* __restrict__ out) {
  int wave = threadIdx.x >> 5;
  int lane = threadIdx.x & 31;
  int tile = blockIdx.x * 4 + wave;          // 1024 tiles = (1024/16) x (256/16)
  int tm = tile >> 4;                        // time-tile  [0,64)
  int tn = tile & 15;                        // hidden-tile [0,16)
  int half = lane >> 4;                      // K-group select
  int lp = lane & 15;

  // A: 16x4 fp32 tile of gathered inputs. lane -> M=lp, VGPR v -> K=half*2+v
  int m = tm * 16 + lp;
  long long rowA = idx[m];
  const float* aptr = emb + rowA * HDIM + half * 2;
  // B: 4x16 fp32 tile of Wih^T. lane -> N=lp, VGPR v -> K=half*2+v
  int n = tn * 16 + lp;
  const float* bptr = Wih + (long)n * HDIM + half * 2;

  v8f c = {};
  for (int k0 = 0; k0 < HDIM; k0 += 4) {
    v2f a; a.x = aptr[k0]; a.y = aptr[k0 + 1];
    v2f b; b.x = bptr[k0]; b.y = bptr[k0 + 1];
    // (neg_a, A, neg_b, B, c_mod, C, reuse_a, reuse_b)
    c = __builtin_amdgcn_wmma_f32_16x16x4_f32(false, a, false, b,
                                              (short)0, c, false, false);
  }

  float bias = bih[n] + bhh[n];
  // C/D layout: VGPR v, lanes0-15 -> M=v ; lanes16-31 -> M=v+8 ; N = lane%16
  for (int v = 0; v < 8; ++v) {
    int M = tm * 16 + half * 8 + v;
    long long r = idx[M];
    out[r * HDIM + n] = c[v] + bias;
  }
}

// ---------------------------------------------------------------------------
// Kernel 3: serial recurrence. One workgroup, 512 threads (16 waves), split-K:
// thread (i, g) owns output element i and K-half g. Each thread keeps its
// 128-float slice of W_hh in registers (32 x float4 = 128 VGPRs -> no spills,
// no VGPR-MSB toggles, 4 waves/SIMD for latency hiding). h is double-buffered
// in LDS (uniform broadcast reads); partials reduced through 1KB of LDS.
// xp[t+1] is prefetched one step ahead so HBM latency hides under the MACs.
// ---------------------------------------------------------------------------
__global__ __launch_bounds__(512) void rnn_seq_kernel(
    const float* __restrict__ Whh, const float* __restrict__ h0,
    const float* __restrict__ times, const long long* __restrict__ idx,
    float* __restrict__ out) {
  __shared__ float sH[2][HDIM];      // double-buffered hidden state
  __shared__ float sPart[HDIM];      // split-K partial sums (from g==1)
  __shared__ int   sIdx[TLEN];       // row indices (fit in int32)
  __shared__ float sScale[TLEN];     // 1/dt + 1 per step
  int tid = threadIdx.x;
  int i = tid & (HDIM - 1);          // output element
  int g = tid >> 8;                  // K-half (uniform per wave)

  // --- this thread's 128-float slice of W_hh row i into registers ---
  float4 w[HDIM / 8];
  const float4* W4 = (const float4*)Whh + (long)i * (HDIM / 4) + g * (HDIM / 8);
#pragma unroll
  for (int k4 = 0; k4 < HDIM / 8; ++k4) w[k4] = W4[k4];

  // --- cache indices and per-step scales in LDS ---
  for (int t = tid; t < TLEN; t += 512) {
    sIdx[t] = (int)idx[t];
    int tp = (t + TLEN - 1) & (TLEN - 1);
    float dt = times[t] - times[tp];
    sScale[t] = 1.0f / dt + 1.0f;
  }
  if (g == 0) sH[0][i] = h0[i];      // initial carry (unscaled)
  __syncthreads();

  float xp_next = (g == 0) ? out[(long)sIdx[0] * HDIM + i] : 0.0f;

  for (int t = 0; t < TLEN; ++t) {
    const float4* sH4 = (const float4*)sH[t & 1] + g * (HDIM / 8);
    float4 a4 = {0.f, 0.f, 0.f, 0.f};            // 4 independent FMA chains
#pragma unroll
    for (int k4 = 0; k4 < HDIM / 8; ++k4) {
      float4 hv = sH4[k4];                       // uniform-address broadcast
      a4.x = fmaf(w[k4].x, hv.x, a4.x);
      a4.y = fmaf(w[k4].y, hv.y, a4.y);
      a4.z = fmaf(w[k4].z, hv.z, a4.z);
      a4.w = fmaf(w[k4].w, hv.w, a4.w);
    }
    float p = (a4.x + a4.y) + (a4.z + a4.w);

    if (g) sPart[i] = p;                         // upper-half partial to LDS
    __syncthreads();

    if (g == 0) {
      float acc = xp_next + p + sPart[i];
      // prefetch next step's staged xp (indices distinct -> row still holds
      // xp); wraps harmlessly at t = TLEN-1
      int rNext = sIdx[(t + 1) & (TLEN - 1)];
      xp_next = out[(long)rNext * HDIM + i];

      // fast tanh via v_exp_f32
      float e2 = __expf(2.0f * acc);
      float hn = (e2 - 1.0f) / (e2 + 1.0f);

      out[(long)sIdx[t] * HDIM + i] = hn;        // emit output row
      sH[(t + 1) & 1][i] = hn * sScale[t];       // next carry (other buffer)
    }
    __syncthreads();
  }
}

// ---------------------------------------------------------------------------
extern "C" void kernel_launch(void* const* d_in, const int* in_sizes, int n_in,
                              void* d_out, int out_size, void* d_ws, size_t ws_size,
                              hipStream_t stream) {
  const float* emb     = (const float*)d_in[0];
  const float* Wih     = (const float*)d_in[1];
  const float* Whh     = (const float*)d_in[2];
  const float* bih     = (const float*)d_in[3];
  const float* bhh     = (const float*)d_in[4];
  const float* h0      = (const float*)d_in[5];
  const float* times   = (const float*)d_in[6];
  const long long* idx = (const long long*)d_in[7];
  float* out = (float*)d_out;

  // 1) bulk copy (HBM-bound)
  long n4 = (long)out_size / 4;
  emb_copy_kernel<<<2048, 256, 0, stream>>>(emb, out, n4);

  // 2) time-parallel input projection via fp32 WMMA; staged into output rows
  xproj_wmma_kernel<<<256, 128, 0, stream>>>(emb, Wih, bih, bhh, idx, out);

  // 3) serial recurrence on one WGP, W_hh register-resident across 512 threads
  rnn_seq_kernel<<<1, 512, 0, stream>>>(Whh, h0, times, idx, out);
}